// SipmSensorResponse_29222957482058
// MI455X (gfx1250) — compile-verified
//
#include <hip/hip_runtime.h>
#include <math.h>

// CDNA5 (gfx1250) implementation of the SiPM sensor-response pipeline.
// Pipeline: h = relu(X@W1+b1) -> resp = exp(h@W2+b2) -> gaussian binning expv
//           -> out[s,t] = sum_k resp[k,s]*expv[k,t]  (f32 GEMM, K=32000)
// The big GEMM runs on V_WMMA_F32_16X16X4_F32. resp is never materialized:
// each workgroup owns a 16-sensor strip and builds its resp panel per-32-k
// block in LDS, so resp is computed exactly once in aggregate.
// Stage-B operands are software-pipelined 2 k-steps ahead (all loads
// unconditional so no phi-copies / masked-load drains appear).
// Workspace: h (16.4MB) + expv (71.7MB) + 4 partial C buffers (19.9MB) ~ 108MB.
// (Pipelined bP2 prefetches may read <1KB past the expv region for waves that
//  never use them — still inside d_ws, values never consumed.)

typedef float v2f __attribute__((ext_vector_type(2)));
typedef float v8f __attribute__((ext_vector_type(8)));

namespace {
constexpr int kBatch  = 64;
constexpr int kEv     = 500;
constexpr int kK      = kBatch * kEv;      // 32000 contraction length
constexpr int kHid    = 128;
constexpr int kS      = 2209;              // 47*47 sensors
constexpr int kSTiles = (kS + 15) / 16;    // 139
constexpr int kSPad   = kSTiles * 16;      // 2224
constexpr int kT      = 550;
constexpr int kTPad   = 560;               // multiple of 16
constexpr int kTTiles = kTPad / 16;        // 35
constexpr int kKB     = 32;                // k-block per iteration
constexpr int kNChunk = 4;                 // split K for occupancy (deterministic partials)
constexpr int kKCh    = kK / kNChunk;      // 8000
constexpr float kNorm = 0.3989422804f * 1.41421356237309515f; // INV_SQRT_2PI/sqrt(0.5)
}

// ---------------- Stage 0: h = relu(X @ W1 + b1)  [32000 x 128] ----------------
__global__ __launch_bounds__(256)
void sipm_h_kernel(const float* __restrict__ x, const float* __restrict__ W1,
                   const float* __restrict__ b1, float* __restrict__ h) {
  int idx = blockIdx.x * 256 + threadIdx.x;
  if (idx >= kK * (kHid / 4)) return;
  int k = idx >> 5;
  int j = (idx & 31) << 2;
  float x0 = x[k * 3 + 0], x1 = x[k * 3 + 1], x2 = x[k * 3 + 2];
  const float4 wa = *(const float4*)(W1 + 0 * kHid + j);
  const float4 wb = *(const float4*)(W1 + 1 * kHid + j);
  const float4 wc = *(const float4*)(W1 + 2 * kHid + j);
  const float4 bb = *(const float4*)(b1 + j);
  float4 r;
  r.x = fmaxf(fmaf(x0, wa.x, fmaf(x1, wb.x, fmaf(x2, wc.x, bb.x))), 0.f);
  r.y = fmaxf(fmaf(x0, wa.y, fmaf(x1, wb.y, fmaf(x2, wc.y, bb.y))), 0.f);
  r.z = fmaxf(fmaf(x0, wa.z, fmaf(x1, wb.z, fmaf(x2, wc.z, bb.z))), 0.f);
  r.w = fmaxf(fmaf(x0, wa.w, fmaf(x1, wb.w, fmaf(x2, wc.w, bb.w))), 0.f);
  *(float4*)(h + (size_t)k * kHid + j) = r;
}

// ---------------- Stage 0b: expv[k][t] gaussian binning  [32000 x 560] ----------
__global__ __launch_bounds__(256)
void sipm_expv_kernel(const float* __restrict__ z, const float* __restrict__ mask,
                      float* __restrict__ expv) {
  int idx = blockIdx.x * 256 + threadIdx.x;
  if (idx >= kK * kTPad) return;
  int k = idx / kTPad;
  int t = idx - k * kTPad;
  float v = 0.f;
  if (t < kT) {
    float d = (float)t - z[k];
    v = expf(-d * d) * mask[k] * kNorm;   // exp(-d^2/(2*0.5)) = exp(-d^2)
  }
  expv[idx] = v;                           // zero padding for t in [550,560)
}

// ---------------- Main fused kernel: resp panel (LDS) + f32 WMMA GEMM -----------
// grid = (139 s-tiles, 4 k-chunks), block = 512 (16 waves).
// Wave w accumulates t-tiles {w, w+16, w+32} of a 16(s) x 560(t) strip.
// Note: w < 16 so tile w+16 (< 32 < 35) always exists; tile w+32 only for w < 3.
__global__ __launch_bounds__(512)
void sipm_gemm_kernel(const float* __restrict__ W2, const float* __restrict__ b2,
                      const float* __restrict__ h, const float* __restrict__ expv,
                      float* __restrict__ part) {
  __shared__ float lds_w2[kHid * 16];   // W2 column strip [128][16]
  __shared__ float lds_resp[kKB * 16];  // resp panel [32 k][16 s]

  const int tid    = threadIdx.x;
  const int stile  = blockIdx.x;
  const int chunk  = blockIdx.y;
  const int s_base = stile * 16;

  // Hoist the W2 strip for this workgroup's sensors into LDS (once).
  for (int r = 0; r < 4; ++r) {
    int idx = tid + r * 512;
    int i = idx >> 4, ss = idx & 15;
    int s = s_base + ss;
    lds_w2[idx] = (s < kS) ? W2[(size_t)i * kS + s] : 0.f;
  }

  // Stage-A thread mapping: thread -> one (k-in-block, s-in-tile) dot product.
  const int kk  = tid >> 4;
  const int ssA = tid & 15;
  const int sA  = s_base + ssA;
  const float accInit = (sA < kS) ? b2[sA] : 0.f;

  // Stage-B (WMMA) mapping per the 32-bit 16x4 A / 4x16 B lane layouts.
  const int w    = tid >> 5;
  const int lane = tid & 31;
  const int m    = lane & 15;          // M (for A) / N (for B) index
  const int kb2  = (lane >> 4) << 1;   // lanes 16-31 hold K=2,3
  const int tb0 = w * 16, tb1 = (w + 16) * 16, tb2 = (w + 32) * 16;
  const bool t2ok = (w + 32) < kTTiles;   // wave-uniform

  v8f acc0 = {}, acc1 = {}, acc2 = {};

  __syncthreads();

  for (int kblk = 0; kblk < kKCh / kKB; ++kblk) {
    const int k0 = chunk * kKCh + kblk * kKB;

    // Stage A: resp[k0+kk][sA] = exp(h[k0+kk,:] . W2[:,sA] + b2[sA])
    {
      float acc = accInit;
      const float4* hrow = (const float4*)(h + (size_t)(k0 + kk) * kHid);
      #pragma unroll
      for (int i4 = 0; i4 < kHid / 4; ++i4) {
        float4 hv = hrow[i4];
        int ib = i4 * 4;
        acc = fmaf(hv.x, lds_w2[(ib + 0) * 16 + ssA], acc);
        acc = fmaf(hv.y, lds_w2[(ib + 1) * 16 + ssA], acc);
        acc = fmaf(hv.z, lds_w2[(ib + 2) * 16 + ssA], acc);
        acc = fmaf(hv.w, lds_w2[(ib + 3) * 16 + ssA], acc);
      }
      lds_resp[kk * 16 + ssA] = expf(acc);
    }
    __syncthreads();

    // Stage B: C[s,t] += A(resp^T) x B(expv) via V_WMMA_F32_16X16X4_F32,
    // with A/B operands software-pipelined 2 k-steps ahead. All operand
    // loads are unconditional (no phis across exec-masked regions); only
    // the acc2 WMMA itself is guarded by the wave-uniform t2ok.
    const float* erow = expv + (size_t)(k0 + kb2) * kTPad + m;
    v2f aP[2], bP0[2], bP1[2], bP2[2];
    #pragma unroll
    for (int p = 0; p < 2; ++p) {
      const int ks = p * 4;
      aP[p].x = lds_resp[(ks + kb2) * 16 + m];
      aP[p].y = lds_resp[(ks + kb2 + 1) * 16 + m];
      const float* e = erow + (size_t)ks * kTPad;
      bP0[p].x = e[tb0]; bP0[p].y = e[kTPad + tb0];
      bP1[p].x = e[tb1]; bP1[p].y = e[kTPad + tb1];
      bP2[p].x = e[tb2]; bP2[p].y = e[kTPad + tb2];
    }
    #pragma unroll
    for (int i = 0; i < kKB / 4; ++i) {
      const int slot = i & 1;
      const v2f a   = aP[slot];
      const v2f b0  = bP0[slot];
      const v2f b1v = bP1[slot];
      const v2f b2v = bP2[slot];
      if (i + 2 < kKB / 4) {          // compile-time after unroll
        const int ks = (i + 2) * 4;
        aP[slot].x = lds_resp[(ks + kb2) * 16 + m];
        aP[slot].y = lds_resp[(ks + kb2 + 1) * 16 + m];
        const float* e = erow + (size_t)ks * kTPad;
        bP0[slot].x = e[tb0]; bP0[slot].y = e[kTPad + tb0];
        bP1[slot].x = e[tb1]; bP1[slot].y = e[kTPad + tb1];
        bP2[slot].x = e[tb2]; bP2[slot].y = e[kTPad + tb2];
      }
      acc0 = __builtin_amdgcn_wmma_f32_16x16x4_f32(false, a, false, b0,
                                                   (short)0, acc0, false, false);
      acc1 = __builtin_amdgcn_wmma_f32_16x16x4_f32(false, a, false, b1v,
                                                   (short)0, acc1, false, false);
      if (t2ok)
        acc2 = __builtin_amdgcn_wmma_f32_16x16x4_f32(false, a, false, b2v,
                                                     (short)0, acc2, false, false);
    }
    __syncthreads();
  }

  // Store C: VGPR r -> rows M=r (lanes 0-15) / M=r+8 (lanes 16-31), N = m.
  float* pbase = part + ((size_t)chunk * kSPad + s_base + ((lane >> 4) << 3)) * kTPad + m;
  #pragma unroll
  for (int r = 0; r < 8; ++r) {
    float* prow = pbase + (size_t)r * kTPad;
    prow[tb0] = acc0[r];
    prow[tb1] = acc1[r];
    if (t2ok) prow[tb2] = acc2[r];
  }
}

// ---------------- Final deterministic reduction over k-chunks -------------------
__global__ __launch_bounds__(256)
void sipm_reduce_kernel(const float* __restrict__ part, float* __restrict__ out) {
  int idx = blockIdx.x * 256 + threadIdx.x;
  if (idx >= kS * kT) return;
  int s = idx / kT, t = idx - s * kT;
  size_t off = (size_t)s * kTPad + t;
  float sum = 0.f;
  #pragma unroll
  for (int c = 0; c < kNChunk; ++c)
    sum += part[(size_t)c * kSPad * kTPad + off];
  out[idx] = sum;
}

extern "C" void kernel_launch(void* const* d_in, const int* in_sizes, int n_in,
                              void* d_out, int out_size, void* d_ws, size_t ws_size,
                              hipStream_t stream) {
  const float* x   = (const float*)d_in[0];  // simulator_input [64,500,3]
  const float* z   = (const float*)d_in[1];  // z_positions    [64,500]
  const float* msk = (const float*)d_in[2];  // mask           [64,500]
  const float* W1  = (const float*)d_in[3];  // [3,128]
  const float* b1  = (const float*)d_in[4];  // [128]
  const float* W2  = (const float*)d_in[5];  // [128,2209]
  const float* b2  = (const float*)d_in[6];  // [2209]
  float* out = (float*)d_out;                // [47,47,550] == [2209,550]

  float* ws   = (float*)d_ws;
  float* h    = ws;                               //  32000*128  f32
  float* expv = h + (size_t)kK * kHid;            //  32000*560  f32
  float* part = expv + (size_t)kK * kTPad;        //  4*2224*560 f32  (~108MB total)

  sipm_h_kernel<<<(kK * (kHid / 4) + 255) / 256, 256, 0, stream>>>(x, W1, b1, h);
  sipm_expv_kernel<<<(kK * kTPad + 255) / 256, 256, 0, stream>>>(z, msk, expv);
  sipm_gemm_kernel<<<dim3(kSTiles, kNChunk), 512, 0, stream>>>(W2, b2, h, expv, part);
  sipm_reduce_kernel<<<(kS * kT + 255) / 256, 256, 0, stream>>>(part, out);
}